// GCNConv_65695819760049
// MI455X (gfx1250) — compile-verified
//
#include <hip/hip_runtime.h>
#include <hip/hip_bf16.h>

typedef __attribute__((ext_vector_type(2))) float v2f;
typedef __attribute__((ext_vector_type(8))) float v8f;

#define NEG_INF (-3.402823466e38f)

// ---------------- workspace layout (in floats) ----------------
// inv_norm : 50048
// ew       : 600064   (edge weights, overwritten with exp() in pass 4)
// pmax     : 75008
// psum     : 1024
// scal     : 128      (scal[0]=gmax, scal[1]=1/sum)
// aggr     : 6400000
#define OFF_INV   0
#define OFF_EW    50048
#define OFF_PMAX  650112
#define OFF_PSUM  725120
#define OFF_SCAL  726144
#define OFF_AGGR  726272

#define N_NODES 50000
#define N_EDGES 600000
#define D_FEAT  128

// ---------------- kernel 1: inverse L2 norms, wave per node ----------------
__global__ __launch_bounds__(256) void k_inv_norm(const float* __restrict__ x,
                                                  float* __restrict__ inv_norm) {
    int wid  = (blockIdx.x * 256 + threadIdx.x) >> 5;   // node id
    int lane = threadIdx.x & 31;
    if (wid >= N_NODES) return;
    const float4* xv = (const float4*)(x + (size_t)wid * D_FEAT);
    float4 v = xv[lane];
    float s = v.x * v.x + v.y * v.y + v.z * v.z + v.w * v.w;
    #pragma unroll
    for (int off = 16; off; off >>= 1) s += __shfl_xor(s, off, 32);
    if (lane == 0) {
        float n = sqrtf(s);
        inv_norm[wid] = 1.0f / fmaxf(n, 1e-12f);
    }
}

// ---------------- kernel 2: edge cosine sim + per-block max ----------------
__global__ __launch_bounds__(256) void k_edge_dot(const float* __restrict__ x,
                                                  const long long* __restrict__ row,
                                                  const long long* __restrict__ col,
                                                  const float* __restrict__ inv_norm,
                                                  float* __restrict__ ew,
                                                  float* __restrict__ pmax) {
    int wid  = (blockIdx.x * 256 + threadIdx.x) >> 5;   // edge id
    int lane = threadIdx.x & 31;
    __shared__ float smax[8];
    float val = NEG_INF;
    if (wid < N_EDGES) {
        long long r = row[wid], c = col[wid];
        const float4* xr = (const float4*)(x + (size_t)r * D_FEAT);
        const float4* xc = (const float4*)(x + (size_t)c * D_FEAT);
        float4 a = xr[lane];
        float4 b = xc[lane];
        float d = a.x * b.x + a.y * b.y + a.z * b.z + a.w * b.w;
        #pragma unroll
        for (int off = 16; off; off >>= 1) d += __shfl_xor(d, off, 32);
        d *= inv_norm[r] * inv_norm[c];
        if (lane == 0) ew[wid] = d;
        val = d;
    }
    if (lane == 0) smax[threadIdx.x >> 5] = val;
    __syncthreads();
    if (threadIdx.x < 8) {
        float m = smax[threadIdx.x];
        #pragma unroll
        for (int off = 4; off; off >>= 1) m = fmaxf(m, __shfl_xor(m, off, 32));
        if (threadIdx.x == 0) pmax[blockIdx.x] = m;
    }
}

// ---------------- kernel 3: global max (single block) ----------------
__global__ __launch_bounds__(1024) void k_reduce_max(const float* __restrict__ pmax,
                                                     int n, float* __restrict__ scal) {
    float m = NEG_INF;
    for (int i = threadIdx.x; i < n; i += 1024) m = fmaxf(m, pmax[i]);
    __shared__ float sm[32];
    #pragma unroll
    for (int off = 16; off; off >>= 1) m = fmaxf(m, __shfl_xor(m, off, 32));
    if ((threadIdx.x & 31) == 0) sm[threadIdx.x >> 5] = m;
    __syncthreads();
    if (threadIdx.x < 32) {
        m = sm[threadIdx.x];
        #pragma unroll
        for (int off = 16; off; off >>= 1) m = fmaxf(m, __shfl_xor(m, off, 32));
        if (threadIdx.x == 0) scal[0] = m;
    }
}

// ---------------- kernel 4: exp(w - gmax) in place + per-block sums --------
__global__ __launch_bounds__(256) void k_exp_partial(float* __restrict__ ew,
                                                     const float* __restrict__ scal,
                                                     float* __restrict__ psum) {
    float gmax = scal[0];
    float s = 0.0f;
    int base = blockIdx.x * 2048 + threadIdx.x;
    #pragma unroll
    for (int j = 0; j < 8; ++j) {
        int i = base + j * 256;
        if (i < N_EDGES) {
            float v = __expf(ew[i] - gmax);
            ew[i] = v;
            s += v;
        }
    }
    __shared__ float sm[8];
    #pragma unroll
    for (int off = 16; off; off >>= 1) s += __shfl_xor(s, off, 32);
    if ((threadIdx.x & 31) == 0) sm[threadIdx.x >> 5] = s;
    __syncthreads();
    if (threadIdx.x < 8) {
        s = sm[threadIdx.x];
        #pragma unroll
        for (int off = 4; off; off >>= 1) s += __shfl_xor(s, off, 32);
        if (threadIdx.x == 0) psum[blockIdx.x] = s;
    }
}

// ---------------- kernel 5: global sum -> 1/sum (single block) -------------
__global__ __launch_bounds__(512) void k_reduce_sum(const float* __restrict__ psum,
                                                    int n, float* __restrict__ scal) {
    float s = 0.0f;
    for (int i = threadIdx.x; i < n; i += 512) s += psum[i];
    __shared__ float sm[16];
    #pragma unroll
    for (int off = 16; off; off >>= 1) s += __shfl_xor(s, off, 32);
    if ((threadIdx.x & 31) == 0) sm[threadIdx.x >> 5] = s;
    __syncthreads();
    if (threadIdx.x < 16) {
        s = sm[threadIdx.x];
        #pragma unroll
        for (int off = 8; off; off >>= 1) s += __shfl_xor(s, off, 32);
        if (threadIdx.x == 0) scal[1] = 1.0f / s;
    }
}

// ---------------- kernel 6: zero the aggregation buffer --------------------
__global__ __launch_bounds__(256) void k_zero(float4* __restrict__ p, int n4) {
    int i = blockIdx.x * 256 + threadIdx.x;
    if (i < n4) p[i] = make_float4(0.f, 0.f, 0.f, 0.f);
}

// ---------------- kernel 7: weighted scatter-add, wave per edge ------------
__global__ __launch_bounds__(256) void k_scatter(const float* __restrict__ x,
                                                 const long long* __restrict__ row,
                                                 const long long* __restrict__ col,
                                                 const float* __restrict__ ew,
                                                 const float* __restrict__ scal,
                                                 float* __restrict__ aggr) {
    int wid  = (blockIdx.x * 256 + threadIdx.x) >> 5;   // edge id
    int lane = threadIdx.x & 31;
    if (wid >= N_EDGES) return;
    long long r = row[wid], c = col[wid];
    float w = ew[wid] * scal[1];
    const float4* xc = (const float4*)(x + (size_t)c * D_FEAT);
    float4 v = xc[lane];
    float* dst = aggr + (size_t)r * D_FEAT + lane * 4;
    atomicAdd(dst + 0, v.x * w);
    atomicAdd(dst + 1, v.y * w);
    atomicAdd(dst + 2, v.z * w);
    atomicAdd(dst + 3, v.w * w);
}

// ---------------- kernel 8: out = aggr @ W^T + b + x via fp32 WMMA ---------
// Block: 256 threads = 8 waves. Each block owns a 16-row slab of nodes
// (staged in LDS); wave j computes the 16x16 output tile at columns [16j,16j+16).
// K-loop: 32 steps of V_WMMA_F32_16X16X4_F32.
// A frag (16x4 f32): lane L -> m = L%16, holds K = {2*(L/16), 2*(L/16)+1}.
// B frag (4x16 f32): lane L -> n = L%16, holds K = {2*(L/16), 2*(L/16)+1};
//   B[k][n] = W[n][k] (since out = aggr @ W^T).
// C/D (16x16 f32, 8 VGPRs): vgpr r, lane L -> m = r + 8*(L/16), n = L%16.
__global__ __launch_bounds__(256) void k_gemm_wmma(const float* __restrict__ aggr,
                                                   const float* __restrict__ Wm,
                                                   const float* __restrict__ bias,
                                                   const float* __restrict__ x,
                                                   float* __restrict__ out) {
    __shared__ float Atile[16 * D_FEAT];   // 8 KB
    int mbase = blockIdx.x * 16;
    int tid   = threadIdx.x;

    // cooperative stage of 16x128 slab of aggr (2048 floats, float4 x2/thread)
    const float4* src = (const float4*)(aggr + (size_t)mbase * D_FEAT);
    float4*       dst = (float4*)Atile;
    #pragma unroll
    for (int i = tid; i < 16 * D_FEAT / 4; i += 256) dst[i] = src[i];
    __syncthreads();

    int wave  = tid >> 5;          // 0..7 -> N tile
    int lane  = tid & 31;
    int nbase = wave * 16;
    int mm    = lane & 15;         // m for A, n for B
    int khalf = lane >> 4;         // which K pair

    const float2* A2 = (const float2*)Atile;                        // [16][64]
    const float2* W2 = (const float2*)(Wm + (size_t)(nbase + mm) * D_FEAT); // row n of W

    v8f acc = {};
    #pragma unroll
    for (int k = 0; k < D_FEAT; k += 4) {
        float2 a2 = A2[mm * (D_FEAT / 2) + (k >> 1) + khalf];
        float2 b2 = W2[(k >> 1) + khalf];
        v2f af, bf;
        af.x = a2.x; af.y = a2.y;
        bf.x = b2.x; bf.y = b2.y;
        acc = __builtin_amdgcn_wmma_f32_16x16x4_f32(
            /*neg_a=*/false, af, /*neg_b=*/false, bf,
            /*c_mod=*/(short)0, acc, /*reuse_a=*/false, /*reuse_b=*/false);
    }

    // epilogue: + bias + residual, store
    int nout  = nbase + mm;
    float bv  = bias[nout];
    int mhalf = khalf * 8;
    #pragma unroll
    for (int r = 0; r < 8; ++r) {
        int m = mbase + mhalf + r;
        size_t idx = (size_t)m * D_FEAT + nout;
        out[idx] = acc[r] + bv + x[idx];
    }
}

// ---------------- launcher -------------------------------------------------
extern "C" void kernel_launch(void* const* d_in, const int* in_sizes, int n_in,
                              void* d_out, int out_size, void* d_ws, size_t ws_size,
                              hipStream_t stream) {
    const float*     x    = (const float*)d_in[0];
    const long long* eidx = (const long long*)d_in[1];   // int64 [2, E]
    const float*     Wm   = (const float*)d_in[2];
    const float*     bias = (const float*)d_in[3];
    float*           out  = (float*)d_out;

    const long long* row = eidx;
    const long long* col = eidx + N_EDGES;

    float* ws       = (float*)d_ws;
    float* inv_norm = ws + OFF_INV;
    float* ew       = ws + OFF_EW;
    float* pmax     = ws + OFF_PMAX;
    float* psum     = ws + OFF_PSUM;
    float* scal     = ws + OFF_SCAL;
    float* aggr     = ws + OFF_AGGR;

    // 1) inverse norms (wave per node)
    k_inv_norm<<<(N_NODES + 7) / 8, 256, 0, stream>>>(x, inv_norm);

    // 2) edge cosine similarities + block-max partials (wave per edge)
    int eblocks = (N_EDGES + 7) / 8;   // 75000
    k_edge_dot<<<eblocks, 256, 0, stream>>>(x, row, col, inv_norm, ew, pmax);

    // 3) global max
    k_reduce_max<<<1, 1024, 0, stream>>>(pmax, eblocks, scal);

    // 4) exp(w - max) + block-sum partials
    int sblocks = (N_EDGES + 2047) / 2048;  // 293
    k_exp_partial<<<sblocks, 256, 0, stream>>>(ew, scal, psum);

    // 5) global sum -> 1/sum
    k_reduce_sum<<<1, 512, 0, stream>>>(psum, sblocks, scal);

    // 6) zero aggr
    int n4 = N_NODES * D_FEAT / 4;
    k_zero<<<(n4 + 255) / 256, 256, 0, stream>>>((float4*)aggr, n4);

    // 7) weighted scatter-add (wave per edge, f32 global atomics)
    k_scatter<<<eblocks, 256, 0, stream>>>(x, row, col, ew, scal, aggr);

    // 8) out = aggr @ W^T + b + x  (fp32 WMMA)
    k_gemm_wmma<<<N_NODES / 16, 256, 0, stream>>>(aggr, Wm, bias, x, out);
}